// ErrorAwareEdgeLoss_66992899883596
// MI455X (gfx1250) — compile-verified
//
#include <hip/hip_runtime.h>
#include <hip/hip_bf16.h>

typedef __attribute__((ext_vector_type(2))) float v2f;
typedef __attribute__((ext_vector_type(8))) float v8f;

// Problem constants (fixed by the reference: h=128, num_logical=64)
#define H   128
#define NL  64
#define LDA 132   // LDS row stride (floats): 132 mod 64 banks = 4 -> conflict-free col reads
#define LDT 132

static __device__ __forceinline__ v8f wmma_f32(v2f a, v2f b, v8f c) {
    // D = A(16x4) * B(4x16) + C(16x16), full fp32
    return __builtin_amdgcn_wmma_f32_16x16x4_f32(false, a, false, b, (short)0, c, false, false);
}

// ---- kernel 0: zero the 64x64 pair-count matrix in workspace ----
__global__ void eal_init_kernel(float* __restrict__ Cw) {
    int idx = blockIdx.x * blockDim.x + threadIdx.x;
    if (idx < NL * NL) Cw[idx] = 0.0f;
}

// ---- kernel 1: histogram the edge pairs (int64) into C ----
__global__ void eal_count_kernel(const long long* __restrict__ pairs,
                                 float* __restrict__ Cw, int n_pairs) {
    int e = blockIdx.x * blockDim.x + threadIdx.x;
    if (e < n_pairs) {
        int i = (int)pairs[2 * e + 0];
        int j = (int)pairs[2 * e + 1];
        atomicAdd(&Cw[i * NL + j], 1.0f);  // equal addends -> order-independent, exact
    }
}

// ---- kernel 2: per-batch  sum_{ij} C[i,j] * (A_b d A_b^T)[i,j]  via f32 WMMA ----
__global__ void __launch_bounds__(256)
eal_main_kernel(const float* __restrict__ P, const float* __restrict__ D,
                const float* __restrict__ Cw, float* __restrict__ partials) {
    __shared__ __align__(16) float sA[NL * LDA];   // A_b: 64 x 128
    __shared__ __align__(16) float sT[32 * LDT];   // T half: 32 x 128
    __shared__ float red[256];

    const int tid  = threadIdx.x;
    const int b    = blockIdx.x;
    const int w    = tid >> 5;        // wave 0..7
    const int lane = tid & 31;
    const int lh   = lane & 15;       // fragment row/col within tile
    const int t    = lane >> 4;       // K-pair selector (0 or 1)

    // ---- stage A: load A_b = P[b, 0:64, 0:128] into LDS ----
    const float* Pb = P + (size_t)b * H * H;
    #pragma unroll
    for (int it = 0; it < 8; ++it) {
        int f   = tid + it * 256;          // float4 index, 2048 total
        int row = f >> 5;                  // 32 float4 per 128-float row
        int c   = (f & 31) << 2;
        float4 v = *(const float4*)(Pb + row * H + c);
        float* dst = &sA[row * LDA + c];
        *(float2*)(dst + 0) = make_float2(v.x, v.y);
        *(float2*)(dst + 2) = make_float2(v.z, v.w);
    }
    __syncthreads();

    float psum = 0.0f;

    #pragma unroll
    for (int h = 0; h < 2; ++h) {
        // ---- step 1: T = A[32h:32h+32, :] @ D ; wave w -> cols [16w, 16w+16) ----
        v8f acc0 = {};
        v8f acc1 = {};
        const int ncol = 16 * w + lh;
        #pragma unroll 4
        for (int k0 = 0; k0 < H; k0 += 4) {
            v2f bf;
            bf.x = D[(k0 + 2 * t + 0) * H + ncol];   // B row K=k0+2t
            bf.y = D[(k0 + 2 * t + 1) * H + ncol];   // B row K=k0+2t+1
            v2f a0 = *(const v2f*)&sA[(32 * h +      lh) * LDA + k0 + 2 * t];
            v2f a1 = *(const v2f*)&sA[(32 * h + 16 + lh) * LDA + k0 + 2 * t];
            acc0 = wmma_f32(a0, bf, acc0);
            acc1 = wmma_f32(a1, bf, acc1);
        }
        // C/D layout: VGPR r holds rows r (lanes 0-15) and r+8 (lanes 16-31)
        #pragma unroll
        for (int r = 0; r < 8; ++r) {
            sT[(     r + 8 * t) * LDT + ncol] = acc0[r];
            sT[(16 + r + 8 * t) * LDT + ncol] = acc1[r];
        }
        __syncthreads();

        // ---- step 2: S_half = T @ A^T ; wave w -> tile (mt = w>>2, nt = w&3) ----
        const int mt = w >> 2, nt = w & 3;
        v8f acc2 = {};
        #pragma unroll 4
        for (int k0 = 0; k0 < H; k0 += 4) {
            v2f a  = *(const v2f*)&sT[(16 * mt + lh) * LDT + k0 + 2 * t];
            // B[k][n] = A[n][k]: same float2-in-K pattern as an A fragment
            v2f bf = *(const v2f*)&sA[(16 * nt + lh) * LDA + k0 + 2 * t];
            acc2 = wmma_f32(a, bf, acc2);
        }
        const int srow = 32 * h + 16 * mt + 8 * t;
        const int scol = 16 * nt + lh;
        #pragma unroll
        for (int r = 0; r < 8; ++r)
            psum += acc2[r] * Cw[(srow + r) * NL + scol];
        __syncthreads();   // protect sT before next half overwrites it
    }

    // ---- deterministic block reduction -> per-batch partial ----
    red[tid] = psum;
    __syncthreads();
    #pragma unroll
    for (int s = 128; s > 0; s >>= 1) {
        if (tid < s) red[tid] += red[tid + s];
        __syncthreads();
    }
    if (tid == 0) partials[b] = red[0];
}

// ---- kernel 3: reduce per-batch partials, apply 1/(batch*n_pairs) ----
__global__ void eal_reduce_kernel(const float* __restrict__ partials, int n,
                                  float* __restrict__ out, float scale) {
    __shared__ float red[256];
    float s = 0.0f;
    for (int i = threadIdx.x; i < n; i += 256) s += partials[i];
    red[threadIdx.x] = s;
    __syncthreads();
    #pragma unroll
    for (int st = 128; st > 0; st >>= 1) {
        if (threadIdx.x < st) red[threadIdx.x] += red[threadIdx.x + st];
        __syncthreads();
    }
    if (threadIdx.x == 0) out[0] = red[0] * scale;
}

extern "C" void kernel_launch(void* const* d_in, const int* in_sizes, int n_in,
                              void* d_out, int out_size, void* d_ws, size_t ws_size,
                              hipStream_t stream) {
    const float*     P     = (const float*)d_in[0];
    const float*     D     = (const float*)d_in[1];
    const long long* pairs = (const long long*)d_in[2];

    const int batch   = in_sizes[0] / (H * H);   // 2048
    const int n_pairs = in_sizes[2] / 2;         // 512

    float* Cw       = (float*)d_ws;              // 64*64 floats
    float* partials = Cw + NL * NL;              // batch floats
    float* out      = (float*)d_out;

    eal_init_kernel<<<(NL * NL + 255) / 256, 256, 0, stream>>>(Cw);
    eal_count_kernel<<<(n_pairs + 255) / 256, 256, 0, stream>>>(pairs, Cw, n_pairs);
    eal_main_kernel<<<batch, 256, 0, stream>>>(P, D, Cw, partials);
    eal_reduce_kernel<<<1, 256, 0, stream>>>(
        partials, batch, out, 1.0f / ((float)batch * (float)n_pairs));
}